// Attention_31885837205937
// MI455X (gfx1250) — compile-verified
//
#include <hip/hip_runtime.h>
#include <stdint.h>

typedef __attribute__((ext_vector_type(2))) float v2f;
typedef __attribute__((ext_vector_type(8))) float v8f;
typedef __attribute__((ext_vector_type(4))) unsigned int v4u;
typedef __attribute__((ext_vector_type(8))) unsigned int v8u;

#define BATCH 8
#define SEQ   2048
#define DIM   512
#define NT    16          // K/V j-tile rows per iteration
#define NTILES (SEQ / NT) // 128
#define KVS   516         // effective LDS row stride (floats): 512 + 2x2 TDM pad
#define SST   17          // padded stride for score-reduction buffer
#define PST   18          // padded (even) stride for P buffer -> aligned b64 loads
#define NEGINF (-1.0e9f)

// One-instruction DMA of a 16x512 f32 tile (row-major, TDM-padded 2 DWORDs per
// 256 DWORDs -> LDS row stride 516) from global memory into LDS.
__device__ __forceinline__ void tdm_load_tile(const float* gsrc, unsigned lds_byte) {
  unsigned long long ga = (unsigned long long)(size_t)gsrc;
  v4u g0;
  g0[0] = 1u;                                        // count=1, user descriptor
  g0[1] = lds_byte;                                  // lds_addr
  g0[2] = (unsigned)(ga & 0xFFFFFFFFu);              // global_addr[31:0]
  g0[3] = (unsigned)((ga >> 32) & 0x01FFFFFFu)       // global_addr[56:32]
        | 0x80000000u;                               // type=2 ("image")
  v8u g1;
  g1[0] = (2u << 16)                                 // data_size = 4B
        | (1u << 20)                                 // pad_enable
        | (7u << 22)                                 // pad_interval = 256 DWORDs
        | (1u << 25);                                // pad_amount   = 2 DWORDs
  g1[1] = ((unsigned)DIM & 0xFFFFu) << 16;           // tensor_dim0 = 512
  g1[2] = ((unsigned)SEQ & 0xFFFFu) << 16;           // tensor_dim1 = 2048
  g1[3] = ((unsigned)DIM & 0xFFFFu) << 16;           // tile_dim0   = 512
  g1[4] = (unsigned)NT;                              // tile_dim1   = 16
  g1[5] = (unsigned)DIM;                             // tensor_dim0_stride = 512
  g1[6] = 0u;
  g1[7] = 0u;
  asm volatile("tensor_load_to_lds %0, %1" :: "s"(g0), "s"(g1) : "memory");
}

__global__ __launch_bounds__(256)
void attn_fa_f32_wmma(const float* __restrict__ x, const int* __restrict__ mask,
                      float* __restrict__ out) {
  extern __shared__ float lds[];
  float* kv0  = lds;                    // tile buffer 0 (also stages Q once)
  float* kv1  = lds + NT * KVS;         // tile buffer 1
  float* sred = kv1 + NT * KVS + 8;     // 16 x SST   cross-wave score reduction
  float* pbuf = sred + 16 * SST;        // 16 x PST   softmax probabilities P
  float* cbuf = pbuf + 16 * PST;        // 16         per-row correction factors
  float* lbuf = cbuf + 16;              // 16         per-row final sums

  const int tid  = threadIdx.x;
  const int wid  = tid >> 5;            // 8 waves
  const int lane = tid & 31;
  const int nl   = lane & 15;           // N / M index within 16
  const int hi   = lane >> 4;           // half-wave select
  const int koff = hi * 2;              // K offset per WMMA f32 layout

  const int bid = blockIdx.x;
  const int b   = bid >> 7;             // SEQ/16 = 128 row-tiles per batch
  const int i0  = (bid & 127) << 4;

  const float* xb = x + (size_t)b * SEQ * DIM;
  const int    d0 = wid * 64;           // this wave's 64-wide D slice
  // TDM inserts 2 pad DWORDs after each 256-DWORD half-row:
  const int    hpad = (d0 >= 256) ? 2 : 0;
  const unsigned kvb0 = (unsigned)(size_t)kv0;
  const unsigned kvb1 = (unsigned)(size_t)kv1;

  // ---- stage Q tile (16 x 512) into LDS via TDM, extract A fragments ----
  if (wid == 0) {
    tdm_load_tile(xb + (size_t)i0 * DIM, kvb0);
    __builtin_amdgcn_s_wait_tensorcnt(0);
  }
  __syncthreads();

  v2f aq[16];                           // Q A-fragments for this wave's D slice
  for (int kc = 0; kc < 16; ++kc)
    aq[kc] = *(const v2f*)(kv0 + hpad + nl * KVS + d0 + 4 * kc + koff);
  const int mrow = mask[b * SEQ + i0 + nl];   // row mask (used by wave 0)
  __syncthreads();

  // pipeline prologue: start DMA of tile 0 into buffer 0
  if (wid == 0)
    tdm_load_tile(xb, kvb0);

  v8f o0 = {0,0,0,0,0,0,0,0}, o1 = {0,0,0,0,0,0,0,0};
  v8f o2 = {0,0,0,0,0,0,0,0}, o3 = {0,0,0,0,0,0,0,0};
  float mrun = -1.0e30f, lrun = 0.0f;   // online-softmax state (wave 0 lanes)

  for (int jt = 0; jt < NTILES; ++jt) {
    const int cur = jt & 1;
    const float* kvh = (cur ? kv1 : kv0) + hpad;

    // ---- issue TDM for the NEXT tile into the other buffer (overlapped) ----
    if (wid == 0 && jt + 1 < NTILES)
      tdm_load_tile(xb + (size_t)(jt + 1) * NT * DIM, cur ? kvb0 : kvb1);

    // overlap: zero reduce buffer while DMA runs
    for (int i = tid; i < 16 * SST; i += 256) sred[i] = 0.0f;

    // wait for CURRENT tile only: TENSORcnt is in-order, <=1 outstanding
    // means the older (current) transfer has completed.
    if (wid == 0) {
      if (jt + 1 < NTILES) __builtin_amdgcn_s_wait_tensorcnt(1);
      else                 __builtin_amdgcn_s_wait_tensorcnt(0);
    }
    __syncthreads();

    // ---- partial scores S[16x16] over this wave's 64-wide D slice ----
    v8f sc = {0,0,0,0,0,0,0,0};
    for (int kc = 0; kc < 16; ++kc) {
      v2f bb = *(const v2f*)(kvh + nl * KVS + d0 + 4 * kc + koff);  // K^T frag
      sc = __builtin_amdgcn_wmma_f32_16x16x4_f32(false, aq[kc], false, bb,
                                                 (short)0, sc, false, false);
    }
    // cross-wave reduction via LDS float atomics
    for (int r = 0; r < 8; ++r)
      atomicAdd(&sred[((hi << 3) + r) * SST + nl], sc[r]);
    __syncthreads();

    // ---- online softmax on the full 16x16 tile (wave 0 only) ----
    if (wid == 0) {
      // lane handles row nl, 8 columns [8*hi, 8*hi+8)
      float sv[8];
      float tmax = -3.0e38f;
      int cbase = hi << 3;
      for (int t = 0; t < 8; ++t) {
        float s = sred[nl * SST + cbase + t];
        if (mrow == 0) s = NEGINF;      // masked row -> uniform softmax
        sv[t] = s;
        tmax = fmaxf(tmax, s);
      }
      tmax = fmaxf(tmax, __shfl_xor(tmax, 16, 32));
      float mnew = fmaxf(mrun, tmax);
      float corr = __expf(mrun - mnew);
      float psum = 0.0f;
      for (int t = 0; t < 8; ++t) {
        float p = __expf(sv[t] - mnew);
        pbuf[nl * PST + cbase + t] = p;
        psum += p;
      }
      psum += __shfl_xor(psum, 16, 32);
      lrun = lrun * corr + psum;
      mrun = mnew;
      if (hi == 0) cbuf[nl] = corr;
    }
    __syncthreads();

    // ---- rescale O and accumulate P @ V over this wave's 64 columns ----
    float cl[8];
    for (int r = 0; r < 8; ++r) cl[r] = cbuf[(hi << 3) + r];
    for (int r = 0; r < 8; ++r) {
      o0[r] *= cl[r]; o1[r] *= cl[r]; o2[r] *= cl[r]; o3[r] *= cl[r];
    }
    for (int kc = 0; kc < 4; ++kc) {
      int k = 4 * kc + koff;
      v2f a = *(const v2f*)(pbuf + nl * PST + k);   // P fragment (aligned b64)
      int col = d0 + nl;
      v2f b0, b1, b2, b3;
      b0.x = kvh[k * KVS + col];      b0.y = kvh[(k + 1) * KVS + col];
      b1.x = kvh[k * KVS + col + 16]; b1.y = kvh[(k + 1) * KVS + col + 16];
      b2.x = kvh[k * KVS + col + 32]; b2.y = kvh[(k + 1) * KVS + col + 32];
      b3.x = kvh[k * KVS + col + 48]; b3.y = kvh[(k + 1) * KVS + col + 48];
      o0 = __builtin_amdgcn_wmma_f32_16x16x4_f32(false, a, false, b0, (short)0, o0, false, false);
      o1 = __builtin_amdgcn_wmma_f32_16x16x4_f32(false, a, false, b1, (short)0, o1, false, false);
      o2 = __builtin_amdgcn_wmma_f32_16x16x4_f32(false, a, false, b2, (short)0, o2, false, false);
      o3 = __builtin_amdgcn_wmma_f32_16x16x4_f32(false, a, false, b3, (short)0, o3, false, false);
    }
    __syncthreads();   // reads of kvh/pbuf/cbuf done before next iteration
  }

  // ---- finalize: divide by row sums and store ----
  if (wid == 0 && hi == 0) lbuf[nl] = lrun;
  __syncthreads();

  float li[8];
  for (int r = 0; r < 8; ++r) li[r] = 1.0f / lbuf[(hi << 3) + r];

  float* ob = out + ((size_t)b * SEQ + i0) * DIM + d0;
  for (int r = 0; r < 8; ++r) {
    int row = (hi << 3) + r;
    ob[(size_t)row * DIM + nl]      = o0[r] * li[r];
    ob[(size_t)row * DIM + 16 + nl] = o1[r] * li[r];
    ob[(size_t)row * DIM + 32 + nl] = o2[r] * li[r];
    ob[(size_t)row * DIM + 48 + nl] = o3[r] * li[r];
  }
}

extern "C" void kernel_launch(void* const* d_in, const int* in_sizes, int n_in,
                              void* d_out, int out_size, void* d_ws, size_t ws_size,
                              hipStream_t stream) {
  (void)in_sizes; (void)n_in; (void)out_size; (void)d_ws; (void)ws_size;
  const float* xp = (const float*)d_in[0];
  const int*   mp = (const int*)d_in[1];
  float*       op = (float*)d_out;

  const size_t shmem = (size_t)(2 * NT * KVS + 8 + 16 * SST + 16 * PST + 32) * sizeof(float);
  // >64KB dynamic LDS: opt in (WGP has 320KB). Not a stream op -> capture-safe.
  (void)hipFuncSetAttribute((const void*)attn_fa_f32_wmma,
                            hipFuncAttributeMaxDynamicSharedMemorySize, (int)shmem);
  dim3 grid(BATCH * (SEQ / 16));   // 1024 workgroups
  dim3 block(256);                 // 8 wave32s
  attn_fa_f32_wmma<<<grid, block, shmem, stream>>>(xp, mp, op);
}